// SparseGuideModel_26594437497073
// MI455X (gfx1250) — compile-verified
//
#include <hip/hip_runtime.h>
#include <math.h>

typedef __attribute__((ext_vector_type(2))) float v2f;
typedef __attribute__((ext_vector_type(8))) float v8f;

#define N_DIM 16384
#define M_DIM 6144
#define K_DIM 128

// ---------------------------------------------------------------------------
// mb_ph[m,k] = mean_beta[m,k] * p_hat[k,m]
// ---------------------------------------------------------------------------
__global__ void mbph_kernel(const float* __restrict__ mean_beta,
                            const float* __restrict__ p_hat,
                            float* __restrict__ mbph) {
    int idx = blockIdx.x * 256 + threadIdx.x;      // over M*K, m-major
    if (idx >= M_DIM * K_DIM) return;
    int m = idx / K_DIM;
    int k = idx % K_DIM;
    mbph[idx] = mean_beta[idx] * p_hat[(size_t)k * M_DIM + m];
}

// ---------------------------------------------------------------------------
// inter_sum[k] = sum_n mean_z[n,k]   (divide by N at use site)
// ---------------------------------------------------------------------------
__global__ void inter_kernel(const float* __restrict__ mean_z,
                             float* __restrict__ inter_sum) {
    int k = threadIdx.x;                            // 128 threads
    int r0 = blockIdx.x * 128;                      // 128 blocks
    float s = 0.f;
    for (int i = 0; i < 128; ++i)
        s += mean_z[(size_t)(r0 + i) * K_DIM + k];
    unsafeAtomicAdd(&inter_sum[k], s);
}

// ---------------------------------------------------------------------------
// GEMM1 (fused): pred = G @ mb_ph ; resid = mean_z - pred - inter ;
//                gsq_diag += column sums of G^2
// One block per 128-row n-tile; 8 waves; wave w owns rows [w*16, w*16+16),
// all 128 k columns (8 WMMA accum tiles).
// B tile is stored K-pair-interleaved: sBi[rowpair][col][2] so each WMMA B
// operand is a single aligned ds_load_b64 (no register shuffles).
// ---------------------------------------------------------------------------
__global__ __launch_bounds__(256) void gemm1_kernel(
    const float* __restrict__ G, const float* __restrict__ mbph,
    const float* __restrict__ mean_z, const float* __restrict__ inter_sum,
    float* __restrict__ resid, float* __restrict__ gsq_diag) {

    __shared__ float sG[128][34];        // G tile: 128 n-rows x 32 m-cols (pad->even stride)
    __shared__ float sBi[16][144][2];    // mb_ph tile: 16 m-rowpairs x 128 k (pad 144)

    const int tid  = threadIdx.x;
    const int lane = tid & 31;
    const int wave = tid >> 5;        // 0..7
    const int n0   = blockIdx.x * 128;
    const int row  = lane & 15;       // WMMA lane row
    const int half = lane >> 4;       // 0: K=0/1, 1: K=2/3

    v8f acc[8] = {};                  // 8 k-tiles of 16x16 f32

    for (int mc = 0; mc < M_DIM; mc += 32) {
        // --- stage G tile; each thread owns one column (tid&31) of the tile
        float colsq = 0.f;
#pragma unroll
        for (int i = 0; i < 16; ++i) {
            int idx = tid + i * 256;          // 4096 elements
            int r = idx >> 5, c = idx & 31;   // c == tid&31 (constant)
            float g = G[(size_t)(n0 + r) * M_DIM + mc + c];
            sG[r][c] = g;
            colsq += g * g;
        }
        unsafeAtomicAdd(&gsq_diag[mc + (tid & 31)], colsq);
        // --- stage mb_ph tile [32 m][128 k], interleaving m-row pairs
#pragma unroll
        for (int i = 0; i < 16; ++i) {
            int idx = tid + i * 256;
            int r = idx >> 7, c = idx & 127;
            sBi[r >> 1][c][r & 1] = mbph[(size_t)(mc + r) * K_DIM + c];
        }
        __syncthreads();

#pragma unroll
        for (int mo = 0; mo < 32; mo += 4) {
            // A: 16(n) x 4(m); pair (mo+2h, mo+2h+1) contiguous in sG row
            v2f a = *(const v2f*)&sG[wave * 16 + row][mo + 2 * half];
            // prefetch all 8 B vectors, then burst 8 WMMAs
            v2f bv[8];
#pragma unroll
            for (int kt = 0; kt < 8; ++kt)
                bv[kt] = *(const v2f*)&sBi[(mo >> 1) + half][kt * 16 + row][0];
#pragma unroll
            for (int kt = 0; kt < 8; ++kt)
                acc[kt] = __builtin_amdgcn_wmma_f32_16x16x4_f32(
                    false, a, false, bv[kt], (short)0, acc[kt], false, false);
        }
        __syncthreads();
    }

    // epilogue: resid = mean_z - pred - inter_sum/N
    const float inv_n = 1.0f / (float)N_DIM;
#pragma unroll
    for (int kt = 0; kt < 8; ++kt) {
#pragma unroll
        for (int v = 0; v < 8; ++v) {
            int r = (lane < 16) ? v : v + 8;        // C layout: VGPR v -> rows v / v+8
            int c = kt * 16 + row;
            size_t n = (size_t)(n0 + wave * 16 + r);
            resid[n * K_DIM + c] =
                mean_z[n * K_DIM + c] - acc[kt][v] - inter_sum[c] * inv_n;
        }
    }
}

// ---------------------------------------------------------------------------
// GEMM2: zkg_acc[k,m] += sum over n-chunk of resid[n,k] * G[n,m]
// Grid (48 m-tiles, 8 n-chunks); wave w owns k rows [w*16, w*16+16),
// 128 m columns (8 WMMA accum tiles). Partials combined via f32 atomics.
// Both tiles stored n-pair-interleaved for single-b64 WMMA operand loads.
// ---------------------------------------------------------------------------
__global__ __launch_bounds__(256) void gemm2_kernel(
    const float* __restrict__ G, const float* __restrict__ resid,
    float* __restrict__ zkg_acc) {

    __shared__ float sGi[16][144][2];    // 16 n-rowpairs x 128 m-cols
    __shared__ float sRi[16][144][2];    // 16 n-rowpairs x 128 k-cols

    const int tid  = threadIdx.x;
    const int lane = tid & 31;
    const int wave = tid >> 5;
    const int m0   = blockIdx.x * 128;
    const int nb   = blockIdx.y * 2048;
    const int row  = lane & 15;
    const int half = lane >> 4;

    v8f acc[8] = {};

    for (int nc = 0; nc < 2048; nc += 32) {
        int n0 = nb + nc;
#pragma unroll
        for (int i = 0; i < 16; ++i) {
            int idx = tid + i * 256;
            int r = idx >> 7, c = idx & 127;
            sGi[r >> 1][c][r & 1] = G[(size_t)(n0 + r) * M_DIM + m0 + c];
            sRi[r >> 1][c][r & 1] = resid[(size_t)(n0 + r) * K_DIM + c];
        }
        __syncthreads();

#pragma unroll
        for (int no = 0; no < 32; no += 4) {
            // A: 16(k) x 4(n) = resid^T, one b64 per lane
            v2f a = *(const v2f*)&sRi[(no >> 1) + half][wave * 16 + row][0];
            v2f bv[8];
#pragma unroll
            for (int mt = 0; mt < 8; ++mt)
                bv[mt] = *(const v2f*)&sGi[(no >> 1) + half][mt * 16 + row][0];
#pragma unroll
            for (int mt = 0; mt < 8; ++mt)
                acc[mt] = __builtin_amdgcn_wmma_f32_16x16x4_f32(
                    false, a, false, bv[mt], (short)0, acc[mt], false, false);
        }
        __syncthreads();
    }

#pragma unroll
    for (int mt = 0; mt < 8; ++mt) {
#pragma unroll
        for (int v = 0; v < 8; ++v) {
            int r = (lane < 16) ? v : v + 8;
            int k = wave * 16 + r;
            int m = m0 + mt * 16 + row;
            unsafeAtomicAdd(&zkg_acc[(size_t)k * M_DIM + m], acc[mt][v]);
        }
    }
}

// ---------------------------------------------------------------------------
// Epilogue: zkg += local^T ; var = 1/(tau+gsq) ; mbn = zkg*var ;
// p_hat_new = clip(sigmoid(logit(p) + 0.5*mbn^2/var)). out = [3,K,M].
// out slot 0 currently holds the zkg accumulation; overwritten in place.
// ---------------------------------------------------------------------------
__global__ void epilogue_kernel(const float* __restrict__ mbph,
                                const float* __restrict__ gsq_diag,
                                const float* __restrict__ tau_beta,
                                const float* __restrict__ p,
                                float* __restrict__ out) {
    int idx = blockIdx.x * 256 + threadIdx.x;       // over K*M, k-major
    if (idx >= K_DIM * M_DIM) return;
    int k = idx / M_DIM;
    int m = idx % M_DIM;
    float g    = gsq_diag[m];
    float zkg  = out[idx] + mbph[(size_t)m * K_DIM + k] * g;   // + local^T
    float var  = 1.0f / (tau_beta[k] + g);
    float mbn  = zkg * var;
    float pk   = p[k];
    float lg   = logf(pk / (1.0f - pk));
    float t    = lg + 0.5f * mbn * mbn / var;
    float ph   = 1.0f / (1.0f + expf(-t));
    ph = fminf(fmaxf(ph, 1e-8f), 1.0f - 1e-8f);
    out[idx]                        = mbn;   // mean_beta_new
    out[K_DIM * M_DIM + idx]        = var;   // var_beta
    out[2 * K_DIM * M_DIM + idx]    = ph;    // p_hat_new
}

// ---------------------------------------------------------------------------
extern "C" void kernel_launch(void* const* d_in, const int* in_sizes, int n_in,
                              void* d_out, int out_size, void* d_ws, size_t ws_size,
                              hipStream_t stream) {
    (void)in_sizes; (void)n_in; (void)out_size; (void)ws_size;
    const float* G         = (const float*)d_in[0];
    const float* mean_z    = (const float*)d_in[1];
    const float* mean_beta = (const float*)d_in[2];
    const float* p_hat     = (const float*)d_in[3];
    const float* tau_beta  = (const float*)d_in[4];
    const float* p         = (const float*)d_in[5];
    float* out = (float*)d_out;

    // workspace layout (floats): mb_ph[M*K] | inter[K] | gsq[M] | resid[N*K]
    float* ws    = (float*)d_ws;
    float* mbph  = ws;
    float* inter = mbph + (size_t)M_DIM * K_DIM;
    float* gsq   = inter + K_DIM;
    float* resid = gsq + M_DIM;

    // zero accumulators: inter+gsq (contiguous) and out slot 0 (zkg acc)
    hipMemsetAsync(inter, 0, (size_t)(K_DIM + M_DIM) * sizeof(float), stream);
    hipMemsetAsync(out, 0, (size_t)K_DIM * M_DIM * sizeof(float), stream);

    mbph_kernel<<<(M_DIM * K_DIM + 255) / 256, 256, 0, stream>>>(mean_beta, p_hat, mbph);
    inter_kernel<<<N_DIM / 128, K_DIM, 0, stream>>>(mean_z, inter);
    gemm1_kernel<<<N_DIM / 128, 256, 0, stream>>>(G, mbph, mean_z, inter, resid, gsq);
    gemm2_kernel<<<dim3(M_DIM / 128, N_DIM / 2048), 256, 0, stream>>>(G, resid, out);
    epilogue_kernel<<<(K_DIM * M_DIM + 255) / 256, 256, 0, stream>>>(mbph, gsq, tau_beta, p, out);
}